// ELM_AE_Spectral_Pool2d_2671469658357
// MI455X (gfx1250) — compile-verified
//
#include <hip/hip_runtime.h>
#include <hip/hip_bf16.h>

// Problem constants from the reference (hard-coded there too):
// B=64, Z=512, H=W=32 -> N=1024, Q=1, LAM=1e-3
#define B_   64
#define Z_   512
#define N_   1024
#define LAM  1e-3f

typedef float v2f __attribute__((ext_vector_type(2)));
typedef float v8f __attribute__((ext_vector_type(8)));
typedef int   i4  __attribute__((ext_vector_type(4)));

#if __has_builtin(__builtin_amdgcn_global_load_async_to_lds_b128) && \
    __has_builtin(__builtin_amdgcn_s_wait_asynccnt)
#define USE_ASYNC_LDS 1
typedef __attribute__((address_space(1))) i4 gi4;   // global int4
typedef __attribute__((address_space(3))) i4 li4;   // LDS int4
#else
#define USE_ASYNC_LDS 0
#endif

// ---------------------------------------------------------------------------
// Pass 1: h[b,n] = sigmoid( sum_z W[z] * x[b,z,n] ),  partial g = sum h^2
// HBM-bound streaming pass: coalesced over n, W cached in LDS,
// global_prefetch_b8 issued ahead of the z-stream to warm L0/L2.
// ---------------------------------------------------------------------------
__global__ __launch_bounds__(256) void elm_h_kernel(const float* __restrict__ x,
                                                    const float* __restrict__ Wh,
                                                    float* __restrict__ h,
                                                    float* __restrict__ gpart)
{
    const int b = blockIdx.y;
    const int n = blockIdx.x * 256 + threadIdx.x;

    __shared__ float Ws[Z_];
    Ws[threadIdx.x]       = Wh[threadIdx.x];
    Ws[threadIdx.x + 256] = Wh[threadIdx.x + 256];
    __syncthreads();

    const float* xp = x + (size_t)b * Z_ * N_ + n;
    float acc = 0.0f;
    #pragma unroll 8
    for (int z = 0; z < Z_; ++z) {
        // speculative prefetch ~64 rows (256KB) ahead; invalid addr is dropped
        __builtin_prefetch(xp + (size_t)(z + 64) * N_, 0, 1);
        acc = fmaf(Ws[z], xp[(size_t)z * N_], acc);
    }
    float hv = 1.0f / (1.0f + __expf(-acc));
    h[(size_t)b * N_ + n] = hv;

    __shared__ float red[256];
    red[threadIdx.x] = hv * hv;
    __syncthreads();
    for (int s = 128; s > 0; s >>= 1) {
        if (threadIdx.x < s) red[threadIdx.x] += red[threadIdx.x + s];
        __syncthreads();
    }
    if (threadIdx.x == 0) gpart[b * 4 + blockIdx.x] = red[0];
}

// ---------------------------------------------------------------------------
// Pass 2: out[b,z] = (sum_n x[b,z,n]*h[b,n]) / g[b]  via V_WMMA_F32_16X16X4_F32
// One wave per (b, 16-row z tile). x is L2-hot after pass 1 (134MB < 192MB L2).
// Chunks (16 rows x 64 cols) are DMA'd global->LDS with
// GLOBAL_LOAD_ASYNC_TO_LDS_B128 (no VGPR data movement), double-buffered
// against the WMMA accumulation; ASYNCcnt waits gate the fragment reads.
// Padded LDS stride 68 floats: conflict-free ds_load_b64 fragments, 16B-aligned
// float4 slots (272B row pitch).
// ---------------------------------------------------------------------------
#define ZT   16
#define CH   64
#define NCH  (N_ / CH)
#define TSTR 68

__global__ __launch_bounds__(32) void elm_beta_kernel(const float* __restrict__ x,
                                                      const float* __restrict__ h,
                                                      const float* __restrict__ gpart,
                                                      float* __restrict__ out)
{
    const int b    = blockIdx.y;
    const int z0   = blockIdx.x * ZT;
    const int lane = threadIdx.x;          // wave32: 0..31

    __shared__ __align__(16) float hs[N_];
    __shared__ __align__(16) float tile[2][ZT * TSTR];

    const float* xrow = x + ((size_t)b * Z_ + z0) * N_;

    const int m    = lane & 15;            // WMMA row (A) / fragment column id
    const int hi   = lane >> 4;            // 0 or 1
    const int m4   = m * 4;                // float4 column within chunk
    const int koff = hi << 1;              // A/B fragment K offset: 0 or 2

    v8f c = {};

#if USE_ASYNC_LDS
    // async-stage h[b,:] (256 float4, 8 per lane), no VGPR data movement
    {
        const float* hp = h + (size_t)b * N_;
        #pragma unroll
        for (int i = 0; i < 8; ++i) {
            int e4 = (i * 32 + lane) * 4;
            __builtin_amdgcn_global_load_async_to_lds_b128(
                (gi4*)(hp + e4), (li4*)(hs + e4), 0, 0);
        }
    }

    // chunk issue: lane owns rows {hi, hi+2, ..., hi+14}, float4 column m
#define ISSUE_CHUNK(cc, bb)                                                    \
    {                                                                          \
        const float* gsrc  = xrow + (size_t)hi * N_ + (cc) * CH + m4;          \
        float*       lbase = &tile[(bb)][hi * TSTR + m4];                      \
        _Pragma("unroll")                                                      \
        for (int i = 0; i < 8; ++i)                                            \
            __builtin_amdgcn_global_load_async_to_lds_b128(                    \
                (gi4*)(gsrc + (size_t)(2 * i) * N_),                           \
                (li4*)(lbase + 2 * i * TSTR), 0, 0);                           \
    }

    ISSUE_CHUNK(0, 0);

    for (int ch = 0; ch < NCH; ++ch) {
        if (ch + 1 < NCH) {
            ISSUE_CHUNK(ch + 1, (ch + 1) & 1);
            __builtin_amdgcn_s_wait_asynccnt(8);   // current chunk (+hs) landed
        } else {
            __builtin_amdgcn_s_wait_asynccnt(0);
        }
        asm volatile("" ::: "memory");             // keep ds_loads below wait

        const int nb  = ch * CH;
        const int buf = ch & 1;
        #pragma unroll
        for (int k4 = 0; k4 < CH / 4; ++k4) {
            int n0 = k4 * 4;
            v2f a = *(const v2f*)&tile[buf][m * TSTR + n0 + koff];
            v2f bb;
            bb.x = hs[nb + n0 + koff];
            bb.y = hs[nb + n0 + koff + 1];
            c = __builtin_amdgcn_wmma_f32_16x16x4_f32(
                    false, a, false, bb, (short)0, c, false, false);
        }
    }
#undef ISSUE_CHUNK
#else
    // fallback: register staging with NAMED float4s (no array -> no scratch spill)
    {
        const float4* hp  = (const float4*)(h + (size_t)b * N_);
        float4*       hsp = (float4*)hs;
        #pragma unroll
        for (int i = 0; i < 8; ++i) hsp[i * 32 + lane] = hp[i * 32 + lane];
    }
    for (int ch = 0; ch < NCH; ++ch) {
        const float* gsrc  = xrow + (size_t)hi * N_ + ch * CH + m4;
        float*       lbase = &tile[0][hi * TSTR + m4];
        float4 q0 = *(const float4*)(gsrc + (size_t) 0 * N_);
        float4 q1 = *(const float4*)(gsrc + (size_t) 2 * N_);
        float4 q2 = *(const float4*)(gsrc + (size_t) 4 * N_);
        float4 q3 = *(const float4*)(gsrc + (size_t) 6 * N_);
        float4 q4 = *(const float4*)(gsrc + (size_t) 8 * N_);
        float4 q5 = *(const float4*)(gsrc + (size_t)10 * N_);
        float4 q6 = *(const float4*)(gsrc + (size_t)12 * N_);
        float4 q7 = *(const float4*)(gsrc + (size_t)14 * N_);
        *(float4*)(lbase +  0 * TSTR) = q0;
        *(float4*)(lbase +  2 * TSTR) = q1;
        *(float4*)(lbase +  4 * TSTR) = q2;
        *(float4*)(lbase +  6 * TSTR) = q3;
        *(float4*)(lbase +  8 * TSTR) = q4;
        *(float4*)(lbase + 10 * TSTR) = q5;
        *(float4*)(lbase + 12 * TSTR) = q6;
        *(float4*)(lbase + 14 * TSTR) = q7;

        const int nb = ch * CH;
        #pragma unroll
        for (int k4 = 0; k4 < CH / 4; ++k4) {
            int n0 = k4 * 4;
            v2f a = *(const v2f*)&tile[0][m * TSTR + n0 + koff];
            v2f bb;
            bb.x = hs[nb + n0 + koff];
            bb.y = hs[nb + n0 + koff + 1];
            c = __builtin_amdgcn_wmma_f32_16x16x4_f32(
                    false, a, false, bb, (short)0, c, false, false);
        }
    }
#endif

    // g[b] in fixed summation order (deterministic), computed by all lanes
    float g = LAM + gpart[b * 4 + 0] + gpart[b * 4 + 1]
                  + gpart[b * 4 + 2] + gpart[b * 4 + 3];
    float inv_g = 1.0f / g;

    // D layout: VGPR j -> M=j (lanes 0-15) / M=j+8 (lanes 16-31), N=lane%16.
    // Column 0 holds the GEMV result: lanes 0 and 16 write 8 rows each.
    if (m == 0) {
        int zb = z0 + (hi << 3);
        #pragma unroll
        for (int j = 0; j < 8; ++j)
            out[(size_t)b * Z_ + zb + j] = c[j] * inv_g;
    }
}

// ---------------------------------------------------------------------------
extern "C" void kernel_launch(void* const* d_in, const int* in_sizes, int n_in,
                              void* d_out, int out_size, void* d_ws, size_t ws_size,
                              hipStream_t stream)
{
    const float* x   = (const float*)d_in[0];  // [B, Z, H, W] f32
    const float* Wh  = (const float*)d_in[1];  // [Q=1, Z]     f32
    float*       out = (float*)d_out;          // [B, Z] f32

    float* h     = (float*)d_ws;               // B*N floats (256 KB)
    float* gpart = h + (size_t)B_ * N_;        // B*4 floats

    dim3 gA(4, B_), bA(256);
    elm_h_kernel<<<gA, bA, 0, stream>>>(x, Wh, h, gpart);

    dim3 gB(Z_ / ZT, B_), bB(32);
    elm_beta_kernel<<<gB, bB, 0, stream>>>(x, h, gpart, out);
}